// TemporalKplanesSE3fields_3298534884036
// MI455X (gfx1250) — compile-verified
//
#include <hip/hip_runtime.h>
#include <hip/hip_bf16.h>
#include <math.h>

typedef __attribute__((ext_vector_type(2))) float v2f;
typedef __attribute__((ext_vector_type(8))) float v8f;

#define FEATC   64
#define HIDDEN  64
#define OUT_DIM 9
#define TILE_PTS 32
#define FSTRIDE 132   // 128 + 4 pad: breaks LDS bank conflicts on A-frag b64 reads
#define HSTRIDE 66    // 64 + 2 pad

// Bilinear sample of one plane for channels (lane) and (lane+32), accumulate.
__device__ __forceinline__ void bil_acc(const float* __restrict__ pl, int H, int W,
                                        float cx, float cy, int lane,
                                        float& a0, float& a1) {
  float xf = (cx + 1.0f) * 0.5f * (float)(W - 1);
  float yf = (cy + 1.0f) * 0.5f * (float)(H - 1);
  float x0f = floorf(xf), y0f = floorf(yf);
  float wx = xf - x0f, wy = yf - y0f;
  int x0 = (int)x0f; x0 = x0 < 0 ? 0 : (x0 > W - 1 ? W - 1 : x0);
  int x1 = (x0 + 1 > W - 1) ? W - 1 : x0 + 1;
  int y0 = (int)y0f; y0 = y0 < 0 ? 0 : (y0 > H - 1 ? H - 1 : y0);
  int y1 = (y0 + 1 > H - 1) ? H - 1 : y0 + 1;
  float w00 = (1.0f - wx) * (1.0f - wy);
  float w01 = wx * (1.0f - wy);
  float w10 = (1.0f - wx) * wy;
  float w11 = wx * wy;
  int hw = H * W;
  const float* pA = pl + lane * hw;
  const float* pB = pl + (lane + 32) * hw;
  int i00 = y0 * W + x0, i01 = y0 * W + x1;
  int i10 = y1 * W + x0, i11 = y1 * W + x1;
  a0 += w00 * pA[i00] + w01 * pA[i01] + w10 * pA[i10] + w11 * pA[i11];
  a1 += w00 * pB[i00] + w01 * pB[i01] + w10 * pB[i10] + w11 * pB[i11];
}

__global__ __launch_bounds__(256) void kplanes_se3_kernel(
    const float* __restrict__ inp, const float* __restrict__ aabb,
    const float* __restrict__ pt0, const float* __restrict__ pt1,
    const float* __restrict__ pt2, const float* __restrict__ pt3,
    const float* __restrict__ pt4, const float* __restrict__ pt5,
    const float* __restrict__ ps0, const float* __restrict__ ps1,
    const float* __restrict__ ps2,
    const float* __restrict__ W1, const float* __restrict__ bias1,
    const float* __restrict__ W2, const float* __restrict__ bias2,
    float* __restrict__ out, int Np) {
  __shared__ float sFeat[TILE_PTS * FSTRIDE];  // 32 x 128 (+pad) feature tile
  __shared__ float sH[TILE_PTS * HSTRIDE];     // 32 x 64 (+pad) hidden tile
  __shared__ float sW2[64 * 16];               // W2 zero-padded 64x16
  __shared__ float sOut[TILE_PTS * 16];        // 32 x 16 output tile

  const int tid  = threadIdx.x;
  const int lane = tid & 31;
  const int wv   = tid >> 5;
  const int pbase = blockIdx.x * TILE_PTS;

  // Stage W2 zero-padded to 64x16 (independent of feat phase).
  for (int i = tid; i < 64 * 16; i += 256) {
    int r = i >> 4, c = i & 15;
    sW2[i] = (c < OUT_DIM) ? W2[r * OUT_DIM + c] : 0.0f;
  }

  const float lo0 = aabb[0], lo1 = aabb[1], lo2 = aabb[2];
  const float hi0 = aabb[3], hi1 = aabb[4], hi2 = aabb[5];

  // ---- Phase 1: K-planes gather. Each wave does 4 points; lanes = channels.
  for (int i = 0; i < 4; ++i) {
    int prow = wv * 4 + i;
    int p = pbase + prow;
    int pp = (p < Np) ? p : (Np - 1);
    float r0 = inp[pp * 4 + 0], r1 = inp[pp * 4 + 1];
    float r2 = inp[pp * 4 + 2], r3 = inp[pp * 4 + 3];
    float c0 = (r0 - lo0) / (hi0 - lo0) * 2.0f - 1.0f;
    float c1 = (r1 - lo1) / (hi1 - lo1) * 2.0f - 1.0f;
    float c2 = (r2 - lo2) / (hi2 - lo2) * 2.0f - 1.0f;
    float c3 = r3 * 2.0f - 1.0f;

    float t0 = 0.0f, t1 = 0.0f, s0 = 0.0f, s1 = 0.0f;
    // planes_t: combs (0,1),(0,2),(0,3),(1,2),(1,3),(2,3); shape (64, RES[c1], RES[c0])
    bil_acc(pt0, 256, 256, c0, c1, lane, t0, t1);
    bil_acc(pt1, 256, 256, c0, c2, lane, t0, t1);
    bil_acc(pt2, 150, 256, c0, c3, lane, t0, t1);
    bil_acc(pt3, 256, 256, c1, c2, lane, t0, t1);
    bil_acc(pt4, 150, 256, c1, c3, lane, t0, t1);
    bil_acc(pt5, 150, 256, c2, c3, lane, t0, t1);
    // planes_s: combs (0,1),(0,2),(1,2); all 256x256
    bil_acc(ps0, 256, 256, c0, c1, lane, s0, s1);
    bil_acc(ps1, 256, 256, c0, c2, lane, s0, s1);
    bil_acc(ps2, 256, 256, c1, c2, lane, s0, s1);

    sFeat[prow * FSTRIDE + lane]      = t0;
    sFeat[prow * FSTRIDE + lane + 32] = t1;
    sFeat[prow * FSTRIDE + 64 + lane] = s0;
    sFeat[prow * FSTRIDE + 96 + lane] = s1;
  }
  __syncthreads();

  // ---- Phase 2: GEMM1  h = relu(feat @ W1 + b1)   via v_wmma_f32_16x16x4_f32
  // 8 waves <-> 2 M-tiles x 4 N-tiles of 16x16; K = 128 in steps of 4.
  {
    const int m = wv & 1;
    const int n = wv >> 1;
    const int row = m * 16 + (lane & 15);
    const int col = n * 16 + (lane & 15);
    const int ko  = (lane >> 4) << 1;  // lanes 0-15: K=k,k+1 ; lanes 16-31: K=k+2,k+3
    v8f acc = {};
#pragma unroll
    for (int k = 0; k < 2 * FEATC; k += 4) {
      v2f a, b;
      a.x = sFeat[row * FSTRIDE + k + ko];
      a.y = sFeat[row * FSTRIDE + k + ko + 1];
      b.x = W1[(k + ko) * HIDDEN + col];       // W1 is (128,64) row-major, L2-resident
      b.y = W1[(k + ko + 1) * HIDDEN + col];
      acc = __builtin_amdgcn_wmma_f32_16x16x4_f32(
          false, a, false, b, (short)0, acc, false, false);
    }
    float bv = bias1[col];
#pragma unroll
    for (int v = 0; v < 8; ++v) {
      int rr = m * 16 + v + ((lane >> 4) << 3);  // C/D layout: M = v (+8 for hi half)
      float hv = acc[v] + bv;
      sH[rr * HSTRIDE + col] = fmaxf(hv, 0.0f);
    }
  }
  __syncthreads();

  // ---- Phase 3: GEMM2  out = h @ W2pad + b2  (waves 0,1; wave-uniform branch)
  if (wv < 2) {
    const int m = wv;
    const int row = m * 16 + (lane & 15);
    const int col = lane & 15;
    const int ko  = (lane >> 4) << 1;
    v8f acc = {};
#pragma unroll
    for (int k = 0; k < HIDDEN; k += 4) {
      v2f a, b;
      a.x = sH[row * HSTRIDE + k + ko];
      a.y = sH[row * HSTRIDE + k + ko + 1];
      b.x = sW2[(k + ko) * 16 + col];
      b.y = sW2[(k + ko + 1) * 16 + col];
      acc = __builtin_amdgcn_wmma_f32_16x16x4_f32(
          false, a, false, b, (short)0, acc, false, false);
    }
    float bv = (col < OUT_DIM) ? bias2[col] : 0.0f;
#pragma unroll
    for (int v = 0; v < 8; ++v) {
      int rr = m * 16 + v + ((lane >> 4) << 3);
      sOut[rr * 16 + col] = acc[v] + bv;
    }
  }
  __syncthreads();

  // ---- Phase 4: 6D -> rotation matrix + translation, write global.
  if (tid < TILE_PTS) {
    int p = pbase + tid;
    if (p < Np) {
      float o[9];
#pragma unroll
      for (int j = 0; j < 9; ++j) o[j] = sOut[tid * 16 + j];
      float a1x = o[0], a1y = o[1], a1z = o[2];
      float a2x = o[3], a2y = o[4], a2z = o[5];
      float n1 = sqrtf(a1x * a1x + a1y * a1y + a1z * a1z);
      n1 = fmaxf(n1, 1e-12f);
      float b1x = a1x / n1, b1y = a1y / n1, b1z = a1z / n1;
      float d = b1x * a2x + b1y * a2y + b1z * a2z;
      float b2x = a2x - d * b1x, b2y = a2y - d * b1y, b2z = a2z - d * b1z;
      float n2 = sqrtf(b2x * b2x + b2y * b2y + b2z * b2z);
      n2 = fmaxf(n2, 1e-12f);
      b2x /= n2; b2y /= n2; b2z /= n2;
      float b3x = b1y * b2z - b1z * b2y;
      float b3y = b1z * b2x - b1x * b2z;
      float b3z = b1x * b2y - b1y * b2x;
      float* Rp = out + (size_t)p * 9;
      Rp[0] = b1x; Rp[1] = b1y; Rp[2] = b1z;
      Rp[3] = b2x; Rp[4] = b2y; Rp[5] = b2z;
      Rp[6] = b3x; Rp[7] = b3y; Rp[8] = b3z;
      float* tp = out + (size_t)Np * 9 + (size_t)p * 3;
      tp[0] = o[6]; tp[1] = o[7]; tp[2] = o[8];
    }
  }
}

extern "C" void kernel_launch(void* const* d_in, const int* in_sizes, int n_in,
                              void* d_out, int out_size, void* d_ws, size_t ws_size,
                              hipStream_t stream) {
  // dict order: inp, aabb, planes_t[6], planes_s[3], W1, b1, W2, b2
  const float* inp  = (const float*)d_in[0];
  const float* aabb = (const float*)d_in[1];
  const float* pt0  = (const float*)d_in[2];
  const float* pt1  = (const float*)d_in[3];
  const float* pt2  = (const float*)d_in[4];
  const float* pt3  = (const float*)d_in[5];
  const float* pt4  = (const float*)d_in[6];
  const float* pt5  = (const float*)d_in[7];
  const float* ps0  = (const float*)d_in[8];
  const float* ps1  = (const float*)d_in[9];
  const float* ps2  = (const float*)d_in[10];
  const float* W1   = (const float*)d_in[11];
  const float* b1   = (const float*)d_in[12];
  const float* W2   = (const float*)d_in[13];
  const float* b2   = (const float*)d_in[14];
  float* out = (float*)d_out;
  int Np = in_sizes[0] / 4;
  int grid = (Np + TILE_PTS - 1) / TILE_PTS;
  kplanes_se3_kernel<<<grid, 256, 0, stream>>>(
      inp, aabb, pt0, pt1, pt2, pt3, pt4, pt5, ps0, ps1, ps2,
      W1, b1, W2, b2, out, Np);
}